// ContentBasedAddressingRead_60352880444050
// MI455X (gfx1250) — compile-verified
//
#include <hip/hip_runtime.h>
#include <math.h>

typedef __attribute__((ext_vector_type(2))) float v2f;
typedef __attribute__((ext_vector_type(8))) float v8f;

#define BB 16
#define NN 32768
#define WW 128
#define RR 8
#define TILES_PER_BATCH (NN / 16)                         // 2048
#define TPW 4                                             // tiles per wave
#define TILES_PER_BLOCK (8 * TPW)                         // 32 tiles / block
#define BLOCKS_PER_BATCH (TILES_PER_BATCH / TILES_PER_BLOCK)  // 64
#define LOG2E 1.44269504088896340736f

// ---------------------------------------------------------------------------
// Kernel 1: WMMA f32 16x16x4 GEMM tiles + memory-row norms + raw scores
//           + per-tile online-softmax partials (max, sum-exp).
// A 16x4 f32 layout: lane L<16 -> (M=L, K=0/1) in v0/v1; L>=16 -> (M=L-16, K=2/3).
// B 4x16: v0/v1 low lanes = K0/K1 (N=lane), high lanes = K2/K3.
// D 16x16: VGPR i: lanes 0-15 -> M=i, lanes 16-31 -> M=i+8; N = lane&15.
// B is tile-invariant -> preloaded to 64 VGPRs. k-outer/tile-inner fully
// unrolled: 4 independent load->wmma chains per k-step keeps many
// global_load_b64 in flight and hides WMMA RAW latency.
// ---------------------------------------------------------------------------
__global__ __launch_bounds__(256)
void cba_scores_kernel(const float* __restrict__ mem,
                       const float* __restrict__ rs,
                       const float* __restrict__ rv,
                       float* __restrict__ scores,   // raw scores -> d_out
                       float* __restrict__ pm,       // per-tile max   [B][R][T]
                       float* __restrict__ ps)       // per-tile sumexp[B][R][T]
{
    __shared__ float rvT[16 * WW];   // read_vectors transposed [col][k]; cols 8..15 = 0
    __shared__ float nv[16];         // ||read_vector_r||; cols 8..15 = 1
    __shared__ float st[16];         // strengths; cols 8..15 = 0

    const int tid = threadIdx.x;
    const int b = blockIdx.x / BLOCKS_PER_BATCH;
    const int tileBase = (blockIdx.x % BLOCKS_PER_BATCH) * TILES_PER_BLOCK;

    // Stage rv (W x R) transposed into LDS as [r][k]; pad columns 8..15 with 0.
    for (int i = tid; i < WW * RR; i += 256) {
        int k = i >> 3, r = i & 7;
        rvT[r * WW + k] = rv[(size_t)b * (WW * RR) + i];
    }
    for (int i = tid; i < WW * RR; i += 256) rvT[WW * RR + i] = 0.f;
    __syncthreads();

    if (tid < 16) {
        float nrmv = 1.f, stv = 0.f;
        if (tid < 8) {
            float s = 0.f;
            for (int k = 0; k < WW; ++k) { float v = rvT[tid * WW + k]; s = fmaf(v, v, s); }
            nrmv = __builtin_amdgcn_sqrtf(s);
            stv = rs[b * RR + tid];
        }
        nv[tid] = nrmv;
        st[tid] = stv;
    }
    __syncthreads();

    const int wave  = tid >> 5;
    const int lane  = tid & 31;
    const int row   = lane & 15;     // A-matrix M row within tile; also B/D column
    const int khalf = lane >> 4;     // 0: K={0,1} mod 4, 1: K={2,3} mod 4
    const int tile0 = tileBase + wave * TPW;

    // Preload the tile-invariant B operand (K=128 -> 32 k-steps x v2f = 64 VGPRs).
    const float* brow = rvT + row * WW + 2 * khalf;
    v2f Breg[32];
#pragma unroll
    for (int k = 0; k < 32; ++k) Breg[k] = *(const v2f*)(brow + 4 * k);

    const float cnv = nv[row];
    const float cst = st[row];

    // Base A pointer for tile t is arow0 + t*16*WW (tiles are 16 consecutive rows).
    const int n0base = tile0 << 4;
    const float* arow0 = mem + ((size_t)b * NN + n0base + row) * WW + 2 * khalf;

    v8f acc[TPW];
    v2f nrmv2[TPW];
#pragma unroll
    for (int t = 0; t < TPW; ++t) { acc[t] = (v8f){}; nrmv2[t] = (v2f){}; }

#pragma unroll
    for (int k = 0; k < 32; ++k) {
#pragma unroll
        for (int t = 0; t < TPW; ++t) {
            v2f A = *(const v2f*)(arow0 + (size_t)t * 16 * WW + 4 * k);
            acc[t] = __builtin_amdgcn_wmma_f32_16x16x4_f32(false, A, false, Breg[k],
                                                           (short)0, acc[t], false, false);
            nrmv2[t] += A * A;   // V_PK_FMA_F32 chain
        }
    }

#pragma unroll
    for (int t = 0; t < TPW; ++t) {
        const int tile = tile0 + t;
        const int n0   = tile << 4;

        float nrm2 = nrmv2[t].x + nrmv2[t].y;
        nrm2 += __shfl_xor(nrm2, 16, 32);   // combine both K-halves of this row

        float sc[8];
        float tmax = -3.402823466e38f;
#pragma unroll
        for (int i = 0; i < 8; ++i) {
            // row M = i + 8*khalf; lane (i + 8*khalf) holds that row's norm^2
            float rnorm2 = __shfl(nrm2, i + 8 * khalf, 32);
            float denom = __builtin_amdgcn_sqrtf(rnorm2) * cnv + 1e-8f;
            float s = cst * acc[t][i] * __builtin_amdgcn_rcpf(denom);
            sc[i] = s;
            tmax = fmaxf(tmax, s);
        }
        tmax = fmaxf(tmax, __shfl_xor(tmax, 16, 32));
        float tsum = 0.f;
#pragma unroll
        for (int i = 0; i < 8; ++i)
            tsum += __builtin_amdgcn_exp2f((sc[i] - tmax) * LOG2E);
        tsum += __shfl_xor(tsum, 16, 32);

        if (row < 8) {
            size_t base = ((size_t)b * NN + n0) * RR + row;
#pragma unroll
            for (int i = 0; i < 8; ++i)
                scores[base + (size_t)(i + 8 * khalf) * RR] = sc[i];
            if (khalf == 0) {
                size_t pidx = ((size_t)b * RR + row) * TILES_PER_BATCH + tile;
                pm[pidx] = tmax;
                ps[pidx] = tsum;
            }
        }
    }
}

// ---------------------------------------------------------------------------
// Kernel 2: merge per-tile partials into global (max, sum) per (b,r).
// ---------------------------------------------------------------------------
__global__ __launch_bounds__(256)
void cba_reduce_kernel(const float* __restrict__ pm, const float* __restrict__ ps,
                       float* __restrict__ Mg, float* __restrict__ Sg)
{
    __shared__ float red[256];
    const int br = blockIdx.x;          // 0..127 = b*8+r
    const int tid = threadIdx.x;
    const float* m = pm + (size_t)br * TILES_PER_BATCH;
    const float* s = ps + (size_t)br * TILES_PER_BATCH;

    float lm = -3.402823466e38f;
    for (int i = tid; i < TILES_PER_BATCH; i += 256) lm = fmaxf(lm, m[i]);
    red[tid] = lm; __syncthreads();
    for (int off = 128; off > 0; off >>= 1) {
        if (tid < off) red[tid] = fmaxf(red[tid], red[tid + off]);
        __syncthreads();
    }
    float gm = red[0];
    __syncthreads();

    float ls = 0.f;
    for (int i = tid; i < TILES_PER_BATCH; i += 256)
        ls += s[i] * __builtin_amdgcn_exp2f((m[i] - gm) * LOG2E);
    red[tid] = ls; __syncthreads();
    for (int off = 128; off > 0; off >>= 1) {
        if (tid < off) red[tid] += red[tid + off];
        __syncthreads();
    }
    if (tid == 0) { Mg[br] = gm; Sg[br] = red[0]; }
}

// ---------------------------------------------------------------------------
// Kernel 3: in-place softmax finalize on d_out.
// ---------------------------------------------------------------------------
__global__ __launch_bounds__(256)
void cba_finalize_kernel(float* __restrict__ out,
                         const float* __restrict__ Mg, const float* __restrict__ Sg)
{
    size_t idx = (size_t)blockIdx.x * 256 + threadIdx.x;   // < B*N*R = 2^22
    int r = (int)(idx & 7);
    int b = (int)(idx >> 18);                              // N*R = 2^18 per batch
    int br = b * 8 + r;
    float inv = __builtin_amdgcn_rcpf(Sg[br]);
    out[idx] = __builtin_amdgcn_exp2f((out[idx] - Mg[br]) * LOG2E) * inv;
}

extern "C" void kernel_launch(void* const* d_in, const int* in_sizes, int n_in,
                              void* d_out, int out_size, void* d_ws, size_t ws_size,
                              hipStream_t stream) {
    const float* mem = (const float*)d_in[0];   // (B, N, W)
    const float* rs  = (const float*)d_in[1];   // (B, R)
    const float* rv  = (const float*)d_in[2];   // (B, W, R)
    float* out = (float*)d_out;                 // (B, N, R)

    // Workspace layout (floats): pm[B*R*T], ps[B*R*T], Mg[128], Sg[128] ~ 2.1 MB
    float* pm = (float*)d_ws;
    float* ps = pm + (size_t)BB * RR * TILES_PER_BATCH;
    float* Mg = ps + (size_t)BB * RR * TILES_PER_BATCH;
    float* Sg = Mg + BB * RR;

    cba_scores_kernel<<<dim3(BB * BLOCKS_PER_BATCH), dim3(256), 0, stream>>>(
        mem, rs, rv, out, pm, ps);
    cba_reduce_kernel<<<dim3(BB * RR), dim3(256), 0, stream>>>(pm, ps, Mg, Sg);
    cba_finalize_kernel<<<dim3((BB * NN * RR) / 256), dim3(256), 0, stream>>>(out, Mg, Sg);
}